// GaussianNet_37709812859496
// MI455X (gfx1250) — compile-verified
//
#include <hip/hip_runtime.h>
#include <hip/hip_bf16.h>
#include <cstdint>

#define EPSV 1e-6f
constexpr int Bb = 16;   // batch
constexpr int Nn = 256;  // nodes
constexpr int Dd = 4;    // pseudo dim
constexpr int Kk = 10;   // kernels
constexpr int Cc = 100;  // channels
constexpr int Cp = 112;  // channels padded to 7 tiles of 16

typedef __attribute__((ext_vector_type(2))) float v2f;
typedef __attribute__((ext_vector_type(8))) float v8f;

// ---------------------------------------------------------------------------
// Stage 1: ag[b,k,n,m] = adj[b,n,m] * exp( sum_d -0.5*(pseudo-mu)^2/(eps+sig^2) )
// One thread per (b,n,m); loops over k. Stored as [B,K,N,N] (m contiguous).
// ---------------------------------------------------------------------------
__global__ void ag_kernel(const float* __restrict__ pseudo,
                          const float* __restrict__ adj,
                          const float* __restrict__ mu,
                          const float* __restrict__ sigma,
                          float* __restrict__ agw) {
    const int idx = blockIdx.x * blockDim.x + threadIdx.x;  // (b*N + n)*N + m
    const int m = idx & (Nn - 1);
    const int n = (idx >> 8) & (Nn - 1);
    const int b = idx >> 16;

    const float4 p = *(const float4*)(pseudo + (size_t)idx * Dd);
    const float a = adj[idx];
    const float pv[4] = {p.x, p.y, p.z, p.w};

#pragma unroll
    for (int k = 0; k < Kk; ++k) {
        float s = 0.0f;
#pragma unroll
        for (int d = 0; d < Dd; ++d) {
            const float sg = sigma[k * Dd + d];
            const float df = pv[d] - mu[k * Dd + d];
            s -= 0.5f * df * df / (EPSV + sg * sg);
        }
        agw[(((size_t)b * Kk + k) * Nn + n) * Nn + m] = a * __expf(s);
    }
}

// ---------------------------------------------------------------------------
// Stage 2: xk[b,k,m,c] = x[b,m,k*C+ :] . W[c,:] + bias[c], zero-padded to Cp.
// One thread per (b,k,m,c). Stored as [B,K,N,Cp] (c contiguous).
// ---------------------------------------------------------------------------
__global__ void xk_kernel(const float* __restrict__ x,
                          const float* __restrict__ W,
                          const float* __restrict__ bias,
                          float* __restrict__ xkw) {
    const int idx = blockIdx.x * blockDim.x + threadIdx.x;  // ((b*K+k)*N + m)*Cp + c
    const int c = idx % Cp;
    const int t = idx / Cp;
    const int m = t & (Nn - 1);
    const int bk = t >> 8;
    const int k = bk % Kk;
    const int b = bk / Kk;

    float s = 0.0f;
    if (c < Cc) {
        const float* xr = x + (size_t)(b * Nn + m) * (Kk * Cc) + k * Cc;
        const float* wr = W + (size_t)c * Cc;
        s = bias[c];
        for (int i = 0; i < Cc; ++i) s = fmaf(xr[i], wr[i], s);
    }
    xkw[idx] = s;
}

// ---------------------------------------------------------------------------
// Stage 3: per (b,k):  D[n,c] = sum_m A[n,m] * Xk[m,c]  via V_WMMA_F32_16X16X4_F32
// One wave per (b,k, n-tile). 7 c-tile accumulators; A fragment reused 7x.
// Fragment layouts per ISA 7.12.2:
//   A (16x4 f32): lane = {M=lane&15}, VGPR j holds K = 2*(lane>>4) + j
//   B (4x16 f32): lane = {N=lane&15}, VGPR j holds K = 2*(lane>>4) + j
//   D (16x16 f32): VGPR r -> row r (lanes 0-15) / row r+8 (lanes 16-31), col=lane&15
// ---------------------------------------------------------------------------
__global__ __launch_bounds__(32) void gemm_kernel(const float* __restrict__ agw,
                                                  const float* __restrict__ xkw,
                                                  float* __restrict__ out) {
    const int lane = threadIdx.x;
    const int lo = lane & 15;
    const int hi = lane >> 4;
    const int nt = blockIdx.x & 15;  // n-tile index (16 tiles of 16 rows)
    const int bk = blockIdx.x >> 4;  // b*K + k
    const int n0 = nt * 16;

    // A: agw[bk][n0+lo][mt + 2*hi + {0,1}]
    const float* Abase = agw + ((size_t)bk * Nn + n0 + lo) * Nn + 2 * hi;
    // B: xkw[bk][mt + 2*hi + {0,1}][ct*16 + lo]
    const float* Bbase = xkw + ((size_t)bk * Nn + 2 * hi) * Cp + lo;

    v8f acc[7] = {};

    for (int mt = 0; mt < Nn; mt += 4) {
        const v2f a = *(const v2f*)(Abase + mt);
        const float* bp = Bbase + (size_t)mt * Cp;
#pragma unroll
        for (int t = 0; t < 7; ++t) {
            v2f bf;
            bf.x = bp[t * 16];
            bf.y = bp[t * 16 + Cp];
            // 8 args: (neg_a, A, neg_b, B, c_mod, C, reuse_a, reuse_b)
            acc[t] = __builtin_amdgcn_wmma_f32_16x16x4_f32(
                false, a, false, bf, (short)0, acc[t], false, false);
        }
    }

    // out[b, k*C + c, n] with c = ct*16+lo (guard c < 100), n = n0 + r + 8*hi
    float* obase = out + (size_t)bk * Cc * Nn;
#pragma unroll
    for (int t = 0; t < 7; ++t) {
        const int c = t * 16 + lo;
        if (c < Cc) {
#pragma unroll
            for (int r = 0; r < 8; ++r) {
                const int n = n0 + r + 8 * hi;
                obase[(size_t)c * Nn + n] = acc[t][r];
            }
        }
    }
}

// ---------------------------------------------------------------------------
extern "C" void kernel_launch(void* const* d_in, const int* in_sizes, int n_in,
                              void* d_out, int out_size, void* d_ws, size_t ws_size,
                              hipStream_t stream) {
    (void)in_sizes; (void)n_in; (void)out_size; (void)ws_size;
    const float* x      = (const float*)d_in[0];
    const float* adj    = (const float*)d_in[1];
    const float* pseudo = (const float*)d_in[2];
    const float* mu     = (const float*)d_in[3];
    const float* sigma  = (const float*)d_in[4];
    const float* W      = (const float*)d_in[5];
    const float* bias   = (const float*)d_in[6];
    float* out = (float*)d_out;

    float* agw = (float*)d_ws;                          // B*K*N*N  floats (41.9 MB)
    float* xkw = agw + (size_t)Bb * Kk * Nn * Nn;       // B*K*N*Cp floats (18.4 MB)

    ag_kernel<<<(Bb * Nn * Nn) / 256, 256, 0, stream>>>(pseudo, adj, mu, sigma, agw);
    xk_kernel<<<(Bb * Kk * Nn * Cp) / 256, 256, 0, stream>>>(x, W, bias, xkw);
    gemm_kernel<<<Bb * Kk * 16, 32, 0, stream>>>(agw, xkw, out);
}